// PairTripletLoss_23931557773832
// MI455X (gfx1250) — compile-verified
//
#include <hip/hip_runtime.h>

// PairTripletLoss on MI455X (gfx1250, wave32).
// Compute-bound fp32 GEMM pair (8.6 GFLOP on 4MB L2-resident data).
// V_WMMA_F32_16X16X4_F32 fed from LDS (conflict-free padded layout), staged
// once per block via async global->LDS copies (ASYNCcnt), fused
// relu/max/sum epilogue.

typedef float v2f __attribute__((ext_vector_type(2)));
typedef float v8f __attribute__((ext_vector_type(8)));

#define NROWS 4096
#define DK    128
#define PTL_MARGIN 0.3f
#define LDS_STRIDE 132  // dwords per row; 132 % 64 == 4 -> conflict-free b64 frag reads

// ---------------------------------------------------------------------------
// Phase A: per-row squared norms of p1/p2, positive-pair distance, and
// zero-init of the rowmax array (harness poisons d_ws each timing pass).
// ---------------------------------------------------------------------------
__global__ __launch_bounds__(256)
void ptl_rownorms(const float* __restrict__ p1, const float* __restrict__ p2,
                  float* __restrict__ rn1, float* __restrict__ rn2,
                  float* __restrict__ dap, unsigned int* __restrict__ rowmax) {
  int i = blockIdx.x * 256 + threadIdx.x;
  if (i >= NROWS) return;
  const float4* a = (const float4*)(p1 + (size_t)i * DK);
  const float4* b = (const float4*)(p2 + (size_t)i * DK);
  float s1 = 0.f, s2 = 0.f, sd = 0.f;
#pragma unroll
  for (int k = 0; k < DK / 4; ++k) {
    float4 x = a[k];
    float4 y = b[k];
    s1 += x.x * x.x + x.y * x.y + x.z * x.z + x.w * x.w;
    s2 += y.x * y.x + y.y * y.y + y.z * y.z + y.w * y.w;
    float dx = x.x - y.x, dy = x.y - y.y, dz = x.z - y.z, dw = x.w - y.w;
    sd += dx * dx + dy * dy + dz * dz + dw * dw;
  }
  rn1[i] = s1;
  rn2[i] = s2;
  dap[i] = sd;
  rowmax[i] = 0u;  // bit pattern of +0.0f; relu outputs are >= 0
}

// ---------------------------------------------------------------------------
// Phase B: fused WMMA GEMM + triplet-loss epilogue.
// Block = 256 threads = 8 waves arranged 4(M) x 2(N); block tile 128x128.
// Full K (=128) staged in LDS once via async global->LDS; wave tile 32x64:
// 2 A frags x 4 B frags -> 8 V_WMMA_F32_16X16X4_F32 per K-step (32 steps).
// blockIdx.z selects the B matrix: 0 -> p1 (D1), 1 -> p2 (D2).
// ---------------------------------------------------------------------------
__global__ __launch_bounds__(256)
void ptl_gemm(const float* __restrict__ p1, const float* __restrict__ p2,
              const float* __restrict__ rn1, const float* __restrict__ rn2,
              const float* __restrict__ dap,
              unsigned int* __restrict__ rowmax,
              float* __restrict__ partials) {
  __shared__ float ldsA[128 * LDS_STRIDE];
  __shared__ float ldsB[128 * LDS_STRIDE];
  __shared__ float wsum[8];

  const int lane = threadIdx.x & 31;
  const int wave = threadIdx.x >> 5;
  const int wm = wave & 3;        // 4 M-slots of 32 rows
  const int wn = wave >> 2;       // 2 N-slots of 64 cols
  const int half = lane >> 4;     // 0: K=0,1 / rows M0..7 ; 1: K=2,3 / rows M8..15
  const int r16 = lane & 15;

  const float* __restrict__ Bp  = blockIdx.z ? p2 : p1;
  const float* __restrict__ rnB = blockIdx.z ? rn2 : rn1;

  const int rowBlk = blockIdx.x * 128;
  const int colBlk = blockIdx.y * 128;

  // ---- one-shot async stage: global -> LDS, 16B per lane, one row per
  // wave-instruction (fully coalesced), no VGPR round-trip.
  {
    const float* gA = p1 + (size_t)rowBlk * DK + (size_t)lane * 4;
    const float* gB = Bp + (size_t)colBlk * DK + (size_t)lane * 4;
#pragma unroll
    for (int j = 0; j < 16; ++j) {
      const int r = wave * 16 + j;
      const float* ga = gA + (size_t)r * DK;
      const float* gb = gB + (size_t)r * DK;
      unsigned la = (unsigned)(size_t)(&ldsA[r * LDS_STRIDE]) + lane * 16;
      unsigned lb = (unsigned)(size_t)(&ldsB[r * LDS_STRIDE]) + lane * 16;
      asm volatile("global_load_async_to_lds_b128 %0, %1, off"
                   :: "v"(la), "v"(ga) : "memory");
      asm volatile("global_load_async_to_lds_b128 %0, %1, off"
                   :: "v"(lb), "v"(gb) : "memory");
    }
    asm volatile("s_wait_asynccnt 0" ::: "memory");
  }
  __syncthreads();

  // ---- GEMM from LDS. Fragment addressing: lane holds row (base + lane%16),
  // K pair 2*(lane/16); stride 132 dwords -> all 32 lanes of a ds_load_b64
  // land on disjoint bank pairs.
  const int aRow = wm * 32 + r16;
  const int bCol = wn * 64 + r16;

  v8f c[2][4];
#pragma unroll
  for (int mi = 0; mi < 2; ++mi)
#pragma unroll
    for (int t = 0; t < 4; ++t) c[mi][t] = (v8f){0.f, 0.f, 0.f, 0.f, 0.f, 0.f, 0.f, 0.f};

#pragma unroll 4
  for (int ks = 0; ks < DK / 4; ++ks) {
    const int ko = ks * 4 + 2 * half;
    v2f a0 = *(const v2f*)&ldsA[aRow * LDS_STRIDE + ko];
    v2f a1 = *(const v2f*)&ldsA[(aRow + 16) * LDS_STRIDE + ko];
    v2f b0 = *(const v2f*)&ldsB[bCol * LDS_STRIDE + ko];
    v2f b1 = *(const v2f*)&ldsB[(bCol + 16) * LDS_STRIDE + ko];
    v2f b2 = *(const v2f*)&ldsB[(bCol + 32) * LDS_STRIDE + ko];
    v2f b3 = *(const v2f*)&ldsB[(bCol + 48) * LDS_STRIDE + ko];
    // 8 independent accumulators between reuses -> covers WMMA RAW window.
    c[0][0] = __builtin_amdgcn_wmma_f32_16x16x4_f32(false, a0, false, b0, (short)0, c[0][0], false, false);
    c[0][1] = __builtin_amdgcn_wmma_f32_16x16x4_f32(false, a0, false, b1, (short)0, c[0][1], false, false);
    c[0][2] = __builtin_amdgcn_wmma_f32_16x16x4_f32(false, a0, false, b2, (short)0, c[0][2], false, false);
    c[0][3] = __builtin_amdgcn_wmma_f32_16x16x4_f32(false, a0, false, b3, (short)0, c[0][3], false, false);
    c[1][0] = __builtin_amdgcn_wmma_f32_16x16x4_f32(false, a1, false, b0, (short)0, c[1][0], false, false);
    c[1][1] = __builtin_amdgcn_wmma_f32_16x16x4_f32(false, a1, false, b1, (short)0, c[1][1], false, false);
    c[1][2] = __builtin_amdgcn_wmma_f32_16x16x4_f32(false, a1, false, b2, (short)0, c[1][2], false, false);
    c[1][3] = __builtin_amdgcn_wmma_f32_16x16x4_f32(false, a1, false, b3, (short)0, c[1][3], false, false);
  }

  // ------------------- fused epilogue -------------------
  // C/D layout: lanes 0-15 hold N=lane, VGPR v = M=v; lanes 16-31 hold
  // N=lane-16, VGPR v = M=v+8.
  const int rowBase = rowBlk + wm * 32;
  const int colBase = colBlk + wn * 64;

  float rnb[4];
  int colt[4];
#pragma unroll
  for (int t = 0; t < 4; ++t) {
    colt[t] = colBase + t * 16 + r16;
    rnb[t] = rnB[colt[t]];
  }

  float lsum = 0.f;
#pragma unroll
  for (int mi = 0; mi < 2; ++mi) {
#pragma unroll
    for (int v = 0; v < 8; ++v) {
      const int row = rowBase + mi * 16 + v + 8 * half;
      // L = relu(dap[row] + M - (rn1[row] + rnb[col] - 2*c))
      const float base = dap[row] + PTL_MARGIN - rn1[row];
      float rmax = 0.f;
#pragma unroll
      for (int t = 0; t < 4; ++t) {
        float l = fmaf(2.0f, c[mi][t][v], base - rnb[t]);
        l = fmaxf(l, 0.0f);
        if (row == colt[t]) l = 0.0f;  // exclude j == i (both planes)
        lsum += l;
        rmax = fmaxf(rmax, l);
      }
      // row max lives across the 16 lanes of this half -> butterfly reduce
      rmax = fmaxf(rmax, __shfl_xor(rmax, 1, 32));
      rmax = fmaxf(rmax, __shfl_xor(rmax, 2, 32));
      rmax = fmaxf(rmax, __shfl_xor(rmax, 4, 32));
      rmax = fmaxf(rmax, __shfl_xor(rmax, 8, 32));
      if (r16 == 0) {
        // non-negative floats: uint order == float order -> exact, deterministic
        atomicMax(rowmax + row, __float_as_uint(rmax));
      }
    }
  }

  // block-level sum -> per-block partial (summed in fixed order later;
  // avoids non-deterministic float atomicAdd)
  lsum += __shfl_xor(lsum, 1, 32);
  lsum += __shfl_xor(lsum, 2, 32);
  lsum += __shfl_xor(lsum, 4, 32);
  lsum += __shfl_xor(lsum, 8, 32);
  lsum += __shfl_xor(lsum, 16, 32);
  if (lane == 0) wsum[wave] = lsum;
  __syncthreads();
  if (threadIdx.x == 0) {
    float s = 0.f;
#pragma unroll
    for (int w = 0; w < 8; ++w) s += wsum[w];
    partials[(blockIdx.z * 32 + blockIdx.y) * 32 + blockIdx.x] = s;
  }
}

// ---------------------------------------------------------------------------
// Phase C: deterministic final reduction -> (sub_loss, all_loss)
// ---------------------------------------------------------------------------
__global__ __launch_bounds__(256)
void ptl_final(const unsigned int* __restrict__ rowmax,
               const float* __restrict__ partials, float* __restrict__ out) {
  __shared__ float red[256];
  const int t = threadIdx.x;

  float s = 0.f;
  for (int j = t; j < NROWS; j += 256) s += __uint_as_float(rowmax[j]);
  red[t] = s;
  __syncthreads();
  for (int off = 128; off > 0; off >>= 1) {
    if (t < off) red[t] += red[t + off];
    __syncthreads();
  }
  const float subsum = red[0];
  __syncthreads();

  float s2 = 0.f;
  for (int j = t; j < 2048; j += 256) s2 += partials[j];
  red[t] = s2;
  __syncthreads();
  for (int off = 128; off > 0; off >>= 1) {
    if (t < off) red[t] += red[t + off];
    __syncthreads();
  }
  if (t == 0) {
    out[0] = subsum / (float)NROWS;                                  // sub_loss
    out[1] = red[0] / (2.0f * (float)NROWS * (float)(NROWS - 1));    // all_loss
  }
}

// ---------------------------------------------------------------------------
extern "C" void kernel_launch(void* const* d_in, const int* in_sizes, int n_in,
                              void* d_out, int out_size, void* d_ws, size_t ws_size,
                              hipStream_t stream) {
  (void)in_sizes; (void)n_in; (void)out_size; (void)ws_size;
  const float* p1 = (const float*)d_in[0];
  const float* p2 = (const float*)d_in[1];
  float* out = (float*)d_out;

  float* ws = (float*)d_ws;
  float* rn1 = ws;                      // [4096]
  float* rn2 = ws + NROWS;              // [4096]
  float* dap = ws + 2 * NROWS;          // [4096]
  unsigned int* rowmax = (unsigned int*)(ws + 3 * NROWS);  // [4096]
  float* partials = ws + 4 * NROWS;     // [2048]

  ptl_rownorms<<<NROWS / 256, 256, 0, stream>>>(p1, p2, rn1, rn2, dap, rowmax);

  dim3 grid(NROWS / 128, NROWS / 128, 2);
  ptl_gemm<<<grid, 256, 0, stream>>>(p1, p2, rn1, rn2, dap, rowmax, partials);

  ptl_final<<<1, 256, 0, stream>>>(rowmax, partials, out);
}